// AA_encoder_40278203302160
// MI455X (gfx1250) — compile-verified
//
#include <hip/hip_runtime.h>
#include <hip/hip_bf16.h>

// ---- static problem config (matches reference) ----
#define NB   64      // graphs
#define LN   128     // nodes per graph
#define HLF  64      // L/2
#define TT   32      // tokens
#define BD   768     // BERT_DIM == HIDDEN_DIM
#define DD   1536    // D
#define NH   8       // heads
#define DKH  192     // D / heads
#define NAA  4096    // B * HALF
#define ATT_SCALE 0.0721687836f   // 1/sqrt(192)

// ---- GEMM tiling ----
#define MT   64      // macro tile M
#define NT   128     // macro tile N
#define KC   32      // K chunk
#define LDSA 34      // A row stride in LDS (32 + 2 pad dwords, via TDM pad)
#define LDSB 132     // B row stride in LDS (128 + 4 pad dwords, via TDM pad)

typedef __attribute__((ext_vector_type(2))) float v2f;
typedef __attribute__((ext_vector_type(8))) float v8f;
typedef __attribute__((ext_vector_type(4))) unsigned int u32x4;
typedef __attribute__((ext_vector_type(8))) int i32x8;
typedef __attribute__((ext_vector_type(4))) int i32x4;

// ---------------------------------------------------------------------------
// Tensor Data Mover: async 2D tile DMA global -> LDS (ISA 08, D# groups 0/1).
//   ldsAddr     : LDS byte offset of tile destination
//   gptr        : global address of tile start
//   tensorD0/D1 : tensor extents (>= tile extents; OOB clip not needed here)
//   tileD0      : contiguous elements per row,  tileD1: rows
//   strideElems : global row stride in elements (4-byte data)
//   padBits     : pad_enable|pad_interval|pad_amount bits for LDS bank padding
// ---------------------------------------------------------------------------
__device__ __forceinline__ void tdm_load_2d(unsigned ldsAddr, const void* gptr,
                                            unsigned tensorD0, unsigned tensorD1,
                                            unsigned tileD0, unsigned tileD1,
                                            unsigned long long strideElems,
                                            unsigned padBits)
{
  const unsigned long long ga = (unsigned long long)(size_t)gptr;
  u32x4 g0;
  g0.x = 1u;                                             // count=1, user mode
  g0.y = ldsAddr;                                        // lds_addr [63:32]
  g0.z = (unsigned)ga;                                   // global_addr lo
  g0.w = ((unsigned)(ga >> 32) & 0x01FFFFFFu) | (2u << 30); // addr[56:32]|type=2
  i32x8 g1;
  g1[0] = (int)((2u << 16) | padBits);                   // data_size=4B + pad cfg
  g1[1] = (int)((tensorD0 & 0xFFFFu) << 16);             // tensor_dim0 [79:48]
  g1[2] = (int)((tensorD0 >> 16) | ((tensorD1 & 0xFFFFu) << 16)); // + dim1
  g1[3] = (int)((tensorD1 >> 16) | (tileD0 << 16));      // tile_dim0 [127:112]
  g1[4] = (int)tileD1;                                   // tile_dim1; tile_dim2=0
  g1[5] = (int)(unsigned)strideElems;                    // dim0_stride lo32
  g1[6] = (int)(unsigned)((strideElems >> 32) & 0xFFFFu);// dim0_stride hi16
  g1[7] = 0;
  const i32x4 z4 = {0, 0, 0, 0};
  const i32x8 z8 = {0, 0, 0, 0, 0, 0, 0, 0};
  __builtin_amdgcn_tensor_load_to_lds(g0, g1, z4, z4, z8, 0);
}

#define PAD_A ((1u << 20) | (4u << 22) | (1u << 25))  // every 32 dw, +2 dw
#define PAD_B ((1u << 20) | (6u << 22) | (3u << 25))  // every 128 dw, +4 dw

// ---------------------------------------------------------------------------
// K1: masked token sum:  S[n,c] = sum_{t<len[n]} bert[n,t,c]   (402 MB stream)
// ---------------------------------------------------------------------------
__global__ void __launch_bounds__(256)
token_sum(const float* __restrict__ bert, const int* __restrict__ blen,
          float* __restrict__ S)
{
  const int n = blockIdx.x;
  const int len = blen[n];
  const float* base = bert + (size_t)n * (TT * BD);
  for (int c = threadIdx.x; c < BD; c += 256) {
    float s = 0.0f;
    for (int t = 0; t < len; ++t) s += base[(size_t)t * BD + c];
    S[(size_t)n * BD + c] = s;
  }
}

// ---------------------------------------------------------------------------
// K2: assemble inner_embed [64,128,1536]
// ---------------------------------------------------------------------------
__global__ void __launch_bounds__(256)
assemble_pooled(const float* __restrict__ pooled, float* __restrict__ embed)
{
  const int idx = blockIdx.x * 256 + threadIdx.x;           // < 4096*768
  const int r = idx / BD, c = idx - r * BD;
  embed[(size_t)((r >> 6) * LN + (r & 63)) * DD + BD + c] = pooled[idx];
}

__global__ void __launch_bounds__(256)
assemble_clause(const float* __restrict__ clause, float* __restrict__ embed)
{
  const int idx = blockIdx.x * 256 + threadIdx.x;           // < 4096*1536
  const int r = idx / DD, c = idx - r * DD;
  embed[(size_t)((r >> 6) * LN + HLF + (r & 63)) * DD + c] = clause[idx];
}

// ---------------------------------------------------------------------------
// K3: fp32 WMMA GEMM  C = A[M,K] @ W[K,N] + biasScale*bias
//   256 threads / 8 waves; block tile 64x128; wave tile 32x32 (2x2 frags);
//   TDM double-buffered LDS staging; V_WMMA_F32_16X16X4_F32 inner loop.
//   inMode : 0 -> A row = m ; 1 -> A rows = b*128+64..127 (clause remap)
//   outMode: 0 -> C row = m ; 1 -> C row = b*128 + (m&63) (AA remap)
// ---------------------------------------------------------------------------
__global__ void __launch_bounds__(256)
gemm_wmma_f32(const float* __restrict__ A, const float* __restrict__ W,
              const float* __restrict__ bias, float biasScale,
              float* __restrict__ C,
              int M, int N, int K, int lda, int ldw, int ldc,
              int inMode, int outMode)
{
  __shared__ float sA[2][MT * LDSA];   // 2 x 8.5 KB
  __shared__ float sB[2][KC * LDSB];   // 2 x 16.5 KB

  const int tid  = threadIdx.x;
  const int lane = tid & 31;
  const int wave = tid >> 5;
  const int wr   = wave >> 2;          // 0..1 -> M half
  const int wc   = wave & 3;           // 0..3 -> N quarter
  const int j    = lane & 15;
  const int h    = lane >> 4;

  const int m0 = blockIdx.y * MT;
  const int n0 = blockIdx.x * NT;

  const int aRowBase = inMode ? ((m0 >> 6) * LN + HLF) : m0;
  const float* aBase = A + (size_t)aRowBase * lda;
  const float* wBase = W + n0;
  const int nChunks = K / KC;

  if (wave == 0) {  // prologue: DMA chunk 0 into buffer 0
    tdm_load_2d((unsigned)(size_t)&sA[0][0], aBase,
                (unsigned)lda, (unsigned)MT, KC, MT,
                (unsigned long long)lda, PAD_A);
    tdm_load_2d((unsigned)(size_t)&sB[0][0], wBase,
                (unsigned)ldw, (unsigned)K, NT, KC,
                (unsigned long long)ldw, PAD_B);
  }

  v8f acc00 = {0.f,0.f,0.f,0.f,0.f,0.f,0.f,0.f};
  v8f acc01 = acc00, acc10 = acc00, acc11 = acc00;

  const int am0 = (wr * 2) * 16, am1 = am0 + 16;   // wave's M tiles in block
  const int bn0 = (wc * 2) * 16, bn1 = bn0 + 16;   // wave's N tiles in block

  for (int c = 0; c < nChunks; ++c) {
    const int buf = c & 1;
    if (wave == 0) {
      if (c + 1 < nChunks) {   // DMA next chunk, then release current one
        const int k0 = (c + 1) * KC;
        tdm_load_2d((unsigned)(size_t)&sA[buf ^ 1][0], aBase + k0,
                    (unsigned)lda, (unsigned)MT, KC, MT,
                    (unsigned long long)lda, PAD_A);
        tdm_load_2d((unsigned)(size_t)&sB[buf ^ 1][0],
                    wBase + (size_t)k0 * ldw,
                    (unsigned)ldw, (unsigned)K, NT, KC,
                    (unsigned long long)ldw, PAD_B);
        __builtin_amdgcn_s_wait_tensorcnt(2);
      } else {
        __builtin_amdgcn_s_wait_tensorcnt(0);
      }
    }
    __syncthreads();

    const float* As = &sA[buf][0];
    const float* Bs = &sB[buf][0];
    #pragma unroll
    for (int kk = 0; kk < KC; kk += 4) {
      const int ka = kk + 2 * h;
      const v2f a0 = *(const v2f*)&As[(am0 + j) * LDSA + ka];
      const v2f a1 = *(const v2f*)&As[(am1 + j) * LDSA + ka];
      v2f b0, b1;
      b0.x = Bs[ka * LDSB + bn0 + j];  b0.y = Bs[(ka + 1) * LDSB + bn0 + j];
      b1.x = Bs[ka * LDSB + bn1 + j];  b1.y = Bs[(ka + 1) * LDSB + bn1 + j];
      acc00 = __builtin_amdgcn_wmma_f32_16x16x4_f32(false, a0, false, b0,
                                                    (short)0, acc00, false, false);
      acc01 = __builtin_amdgcn_wmma_f32_16x16x4_f32(false, a0, false, b1,
                                                    (short)0, acc01, false, false);
      acc10 = __builtin_amdgcn_wmma_f32_16x16x4_f32(false, a1, false, b0,
                                                    (short)0, acc10, false, false);
      acc11 = __builtin_amdgcn_wmma_f32_16x16x4_f32(false, a1, false, b1,
                                                    (short)0, acc11, false, false);
    }
    __syncthreads();
  }

  // epilogue: bias + (optional) AA row remap + store
  const float bv0 = bias ? bias[n0 + bn0 + j] * biasScale : 0.0f;
  const float bv1 = bias ? bias[n0 + bn1 + j] * biasScale : 0.0f;
  auto storeTile = [&](const v8f& acc, int mt, int nt, float bv) {
    #pragma unroll
    for (int v = 0; v < 8; ++v) {
      const int m = m0 + mt + v + 8 * h;
      const size_t crow = outMode ? (size_t)((m >> 6) * LN + (m & 63))
                                  : (size_t)m;
      C[crow * (size_t)ldc + (size_t)(n0 + nt + j)] = acc[v] + bv;
    }
  };
  storeTile(acc00, am0, bn0, bv0);
  storeTile(acc01, am0, bn1, bv1);
  storeTile(acc10, am1, bn0, bv0);
  storeTile(acc11, am1, bn1, bv1);
}

// ---------------------------------------------------------------------------
// K4: per clause-row attention: 8-head masked softmax, head-mean, gating,
//     top-3 selection (jax tie-break: lowest index), denom.
// ---------------------------------------------------------------------------
__global__ void __launch_bounds__(128)
attn_topk(const float* __restrict__ Q, const float* __restrict__ Kb,
          const int* __restrict__ glen,
          float* __restrict__ topv, int* __restrict__ topi,
          float* __restrict__ denomOut)
{
  __shared__ float sQ[DD];
  __shared__ float red[128];
  __shared__ int  redi[128];
  const int r  = blockIdx.x;
  const int b  = r >> 6;
  const int qi = HLF + (r & 63);
  const int t  = threadIdx.x;
  const int len = glen[b];

  const float* qrow = Q + (size_t)r * DD;
  for (int c = t; c < DD; c += 128) sQ[c] = qrow[c];
  __syncthreads();

  const float* krow = Kb + (size_t)(b * LN + t) * DD;
  const bool kv = (t < len);
  float sc[NH];
  #pragma unroll
  for (int hh = 0; hh < NH; ++hh) {
    const float* qh = sQ + hh * DKH;
    const float* kh = krow + hh * DKH;
    float acc = 0.0f;
    #pragma unroll 4
    for (int d = 0; d < DKH; ++d) acc = fmaf(qh[d], kh[d], acc);
    sc[hh] = kv ? acc * ATT_SCALE : -1e9f;
  }

  float attnMean = 0.0f;
  for (int hh = 0; hh < NH; ++hh) {
    red[t] = sc[hh]; __syncthreads();
    for (int s = 64; s > 0; s >>= 1) {
      if (t < s) red[t] = fmaxf(red[t], red[t + s]);
      __syncthreads();
    }
    const float m = red[0]; __syncthreads();
    const float e = __expf(sc[hh] - m);
    red[t] = e; __syncthreads();
    for (int s = 64; s > 0; s >>= 1) {
      if (t < s) red[t] += red[t + s];
      __syncthreads();
    }
    const float ssum = red[0]; __syncthreads();
    attnMean += e / ssum;
  }
  attnMean *= (1.0f / NH);

  const float adjv = ((qi < len) && kv) ? attnMean : 0.0f;

  float remain = adjv;
  float sum3 = 0.0f;
  for (int it = 0; it < 3; ++it) {
    red[t] = remain; redi[t] = t; __syncthreads();
    for (int s = 64; s > 0; s >>= 1) {
      if (t < s) {
        const float v2 = red[t + s]; const int i2 = redi[t + s];
        if (v2 > red[t] || (v2 == red[t] && i2 < redi[t])) {
          red[t] = v2; redi[t] = i2;
        }
      }
      __syncthreads();
    }
    const float wv = red[0];
    const int   wi = redi[0];
    __syncthreads();
    sum3 += wv;
    if (t == 0) { topv[r * 3 + it] = wv; topi[r * 3 + it] = wi; }
    if (t == wi) remain = -3.4e38f;
  }
  if (t == 0) denomOut[r] = sum3 + 1.0f;
}

// ---------------------------------------------------------------------------
// K5: sparse Ax: adjacency has <=3 nonzeros per selected row
// ---------------------------------------------------------------------------
__global__ void __launch_bounds__(256)
sparse_ax(const float* __restrict__ embed, const float* __restrict__ topv,
          const int* __restrict__ topi, float* __restrict__ Ax)
{
  const int r = blockIdx.x;            // 0..4095 (compact clause row)
  const int b = r >> 6;
  const float v0 = topv[r * 3 + 0];
  const float v1 = topv[r * 3 + 1];
  const float v2 = topv[r * 3 + 2];
  const float* e0 = embed + (size_t)(b * LN + topi[r * 3 + 0]) * DD;
  const float* e1 = embed + (size_t)(b * LN + topi[r * 3 + 1]) * DD;
  const float* e2 = embed + (size_t)(b * LN + topi[r * 3 + 2]) * DD;
  float* out = Ax + (size_t)r * DD;
  for (int c = threadIdx.x; c < DD; c += 256)
    out[c] = fmaf(v0, e0[c], fmaf(v1, e1[c], v2 * e2[c]));
}

// ---------------------------------------------------------------------------
// K6: finalize: relu((Ax@W + b)/denom), select vs clause by graph length
// ---------------------------------------------------------------------------
__global__ void __launch_bounds__(256)
finalize(const float* __restrict__ gcnraw, const float* __restrict__ denom,
         const float* __restrict__ clause, const int* __restrict__ glen,
         float* __restrict__ out)
{
  const int idx = blockIdx.x * 256 + threadIdx.x;   // < 4096*1536
  const int r = idx / DD;
  const int b = r >> 6;
  const float x = fmaxf(gcnraw[idx] / denom[r], 0.0f);
  out[idx] = (glen[b] > 1) ? x : clause[idx];
}

// ---------------------------------------------------------------------------
extern "C" void kernel_launch(void* const* d_in, const int* in_sizes, int n_in,
                              void* d_out, int out_size, void* d_ws, size_t ws_size,
                              hipStream_t stream)
{
  (void)in_sizes; (void)n_in; (void)out_size; (void)ws_size;

  const float* bert   = (const float*)d_in[0];
  const float* pooled = (const float*)d_in[1];
  const float* clause = (const float*)d_in[2];
  const int*   blen   = (const int*)d_in[3];
  const int*   glen   = (const int*)d_in[4];
  const float* denseW = (const float*)d_in[9];
  const float* denseB = (const float*)d_in[10];
  const float* Wq     = (const float*)d_in[11];
  const float* bq     = (const float*)d_in[12];
  const float* Wk     = (const float*)d_in[13];
  const float* bk     = (const float*)d_in[14];
  const float* gcnW   = (const float*)d_in[15];
  const float* gcnB   = (const float*)d_in[16];
  float* out = (float*)d_out;

  float* ws    = (float*)d_ws;
  float* S     = ws;                               // 4096*768
  float* embed = S + (size_t)NAA * BD;             // 8192*1536
  float* qbuf  = embed + (size_t)NB * LN * DD;     // 4096*1536 (clause rows)
  float* kbuf  = qbuf + (size_t)NAA * DD;          // 8192*1536
  float* topv  = kbuf + (size_t)NB * LN * DD;      // 4096*3
  int*   topi  = (int*)(topv + (size_t)NAA * 3);   // 4096*3
  float* denom = (float*)(topi + (size_t)NAA * 3); // 4096
  float* Ax    = kbuf;   // reuse: k dead after attention
  float* graw  = qbuf;   // reuse: q dead after attention

  token_sum<<<NAA, 256, 0, stream>>>(bert, blen, S);
  assemble_pooled<<<(NAA * BD) / 256, 256, 0, stream>>>(pooled, embed);
  assemble_clause<<<(NAA * DD) / 256, 256, 0, stream>>>(clause, embed);

  // dense: S[4096,768] @ denseW + 32*b -> embed AA rows, cols [0,768)
  gemm_wmma_f32<<<dim3(BD / NT, NAA / MT), 256, 0, stream>>>(
      S, denseW, denseB, (float)TT, embed, NAA, BD, BD, BD, BD, DD, 0, 1);

  // q projection (clause rows only)
  gemm_wmma_f32<<<dim3(DD / NT, NAA / MT), 256, 0, stream>>>(
      embed, Wq, bq, 1.0f, qbuf, NAA, DD, DD, DD, DD, DD, 1, 0);

  // k projection (all rows)
  gemm_wmma_f32<<<dim3(DD / NT, (NB * LN) / MT), 256, 0, stream>>>(
      embed, Wk, bk, 1.0f, kbuf, NB * LN, DD, DD, DD, DD, DD, 0, 0);

  attn_topk<<<NAA, 128, 0, stream>>>(qbuf, kbuf, glen, topv, topi, denom);
  sparse_ax<<<NAA, 256, 0, stream>>>(embed, topv, topi, Ax);

  // GCN GEMM
  gemm_wmma_f32<<<dim3(DD / NT, NAA / MT), 256, 0, stream>>>(
      Ax, gcnW, gcnB, 1.0f, graw, NAA, DD, DD, DD, DD, DD, 0, 0);

  finalize<<<(NAA * DD) / 256, 256, 0, stream>>>(graw, denom, clause, glen, out);
}